// GNN_Net_30236569764544
// MI455X (gfx1250) — compile-verified
//
#include <hip/hip_runtime.h>
#include <hip/hip_bf16.h>

typedef __attribute__((ext_vector_type(2))) float v2f;
typedef __attribute__((ext_vector_type(8))) float v8f;

#define LEAKY 0.01f
#define BN_EPS 1e-5f

// ---------------------------------------------------------------------------
// WMMA GEMM: C[M x N] = A[M x K] @ B[K x N], row-major fp32.
// One wave per 16x16 tile, V_WMMA_F32_16X16X4_F32, K compile-time (% 4 == 0),
// M % 16 == 0, N % 16 == 0. Fully unrolled: no guards, vector A loads.
// A-frag (32-bit A 16x4): lanes 0-15 -> K=k+0,k+1 ; lanes 16-31 -> K=k+2,k+3.
// C/D layout: vgpr r -> row (r + 8*half), col = lane & 15.
// ---------------------------------------------------------------------------
template <int K>
__global__ __launch_bounds__(32) void wmma_gemm_k(
    const float* __restrict__ A, const float* __restrict__ B,
    float* __restrict__ C, int M, int N)
{
    const int ntn  = N >> 4;
    const int tm   = blockIdx.x / ntn;
    const int tn   = blockIdx.x % ntn;
    const int lane = threadIdx.x & 31;
    const int half = lane >> 4;
    const int lr   = lane & 15;

    const int row = tm * 16 + lr;
    const int col = tn * 16 + lr;
    const float* __restrict__ Arow = A + (size_t)row * K + 2 * half;
    const float* __restrict__ Bcol = B + (size_t)(2 * half) * N + col;

    v8f acc = {};
#pragma unroll
    for (int k = 0; k < K; k += 4) {
        v2f a = *(const v2f*)(Arow + k);           // 8-byte aligned (K%4==0, even off)
        v2f b;
        b.x = Bcol[(size_t)k * N];
        b.y = Bcol[(size_t)k * N + N];
        acc = __builtin_amdgcn_wmma_f32_16x16x4_f32(
            false, a, false, b, (short)0, acc, false, false);
    }
#pragma unroll
    for (int r = 0; r < 8; ++r)
        C[(size_t)(tm * 16 + r + 8 * half) * N + col] = acc[r];
}

// ---------------------------------------------------------------------------
__global__ void zero_f32(float* p, int n) {
    for (int i = blockIdx.x * blockDim.x + threadIdx.x; i < n;
         i += gridDim.x * blockDim.x) p[i] = 0.0f;
}

__global__ void fill_f32(float* p, float v, int n) {
    int i = blockIdx.x * blockDim.x + threadIdx.x;
    if (i < n) p[i] = v;
}

// pad x [n,3] -> xp [n,4] (last col 0)
__global__ void pad_x(const float* __restrict__ x, float* __restrict__ xp, int n) {
    int i = blockIdx.x * blockDim.x + threadIdx.x;
    if (i >= n) return;
    float4 v;
    v.x = x[i * 3 + 0]; v.y = x[i * 3 + 1]; v.z = x[i * 3 + 2]; v.w = 0.0f;
    ((float4*)xp)[i] = v;
}

// pad W0 [3,16] -> [4,16] (zero last row)
__global__ void pad_w0(const float* __restrict__ w, float* __restrict__ wp) {
    int i = threadIdx.x;   // 64 threads
    wp[i] = (i < 48) ? w[i] : 0.0f;
}

// degree count (dst occurrences); deg pre-filled with 1.0 for self-loop
__global__ void count_deg(const int* __restrict__ dst, float* deg, int E) {
    int e = blockIdx.x * blockDim.x + threadIdx.x;
    if (e < E) atomicAdd(&deg[dst[e]], 1.0f);
}

__global__ void finalize_dinv(float* deg, int n) {
    int i = blockIdx.x * blockDim.x + threadIdx.x;
    if (i < n) deg[i] = rsqrtf(deg[i]);   // deg >= 1 always
}

// message scatter, 4 channels per thread:
// out[dst] += hw[src] * dinv[src]*dinv[dst]
__global__ void edge_scatter4(const float* __restrict__ hw,
                              const int* __restrict__ src,
                              const int* __restrict__ dst,
                              const float* __restrict__ dinv,
                              float* out, int E, int dq /* d/4 */)
{
    int idx = blockIdx.x * blockDim.x + threadIdx.x;
    if (idx >= E * dq) return;
    int e = idx / dq, q = idx - e * dq;
    int s = src[e], t = dst[e];
    float w = dinv[s] * dinv[t];
    float4 v = ((const float4*)hw)[(size_t)s * dq + q];
    float* o = out + ((size_t)t * dq + q) * 4;
    atomicAdd(o + 0, v.x * w);
    atomicAdd(o + 1, v.y * w);
    atomicAdd(o + 2, v.z * w);
    atomicAdd(o + 3, v.w * w);
}

// self-loop contribution (after scatter, same stream => ordered, plain add)
__global__ void self_loop_add4(const float* __restrict__ hw,
                               const float* __restrict__ dinv,
                               float* out, int n, int dq)
{
    int idx = blockIdx.x * blockDim.x + threadIdx.x;
    if (idx >= n * dq) return;
    int i = idx / dq;
    float w = dinv[i] * dinv[i];
    float4 v = ((const float4*)hw)[idx];
    float4 o = ((float4*)out)[idx];
    o.x += v.x * w; o.y += v.y * w; o.z += v.z * w; o.w += v.w * w;
    ((float4*)out)[idx] = o;
}

// per-channel sum / sum-of-squares via LDS partials (d <= 128)
__global__ void bn_stats(const float* __restrict__ h, float* sum, float* sumsq,
                         int n, int d)
{
    __shared__ float ssum[128];
    __shared__ float ssq[128];
    for (int c = threadIdx.x; c < d; c += blockDim.x) { ssum[c] = 0.f; ssq[c] = 0.f; }
    __syncthreads();
    const int total = n * d;
    for (int idx = blockIdx.x * blockDim.x + threadIdx.x; idx < total;
         idx += gridDim.x * blockDim.x) {
        int c = idx % d;
        float v = h[idx];
        atomicAdd(&ssum[c], v);
        atomicAdd(&ssq[c], v * v);
    }
    __syncthreads();
    for (int c = threadIdx.x; c < d; c += blockDim.x) {
        atomicAdd(&sum[c], ssum[c]);
        atomicAdd(&sumsq[c], ssq[c]);
    }
}

__global__ void bn_finalize(const float* sum, const float* sumsq,
                            const float* __restrict__ g, const float* __restrict__ b,
                            float* scale, float* shift, int n, int d)
{
    int c = blockIdx.x * blockDim.x + threadIdx.x;
    if (c >= d) return;
    float inv_n = 1.0f / (float)n;
    float mu  = sum[c] * inv_n;
    float var = sumsq[c] * inv_n - mu * mu;
    float s = g[c] * rsqrtf(var + BN_EPS);
    scale[c] = s;
    shift[c] = b[c] - mu * s;
}

__global__ void bn_apply_leaky(float* h, const float* __restrict__ scale,
                               const float* __restrict__ shift, int total, int d)
{
    int idx = blockIdx.x * blockDim.x + threadIdx.x;
    if (idx >= total) return;
    int c = idx % d;
    float v = h[idx] * scale[c] + shift[c];
    h[idx] = (v > 0.f) ? v : LEAKY * v;
}

// scatter_mean pooling
__global__ void pool_scatter(const float* __restrict__ h, const int* __restrict__ batch,
                             float* sums, float* cnt, int n, int d)
{
    int idx = blockIdx.x * blockDim.x + threadIdx.x;
    if (idx >= n * d) return;
    int i = idx / d, c = idx - i * d;
    int bg = batch[i];
    atomicAdd(&sums[bg * d + c], h[idx]);
    if (c == 0) atomicAdd(&cnt[bg], 1.0f);
}

__global__ void pool_div(float* sums, const float* cnt, int total, int d) {
    int idx = blockIdx.x * blockDim.x + threadIdx.x;
    if (idx >= total) return;
    sums[idx] = sums[idx] / fmaxf(cnt[idx / d], 1.0f);
}

// final small linear with bias: out[M x N] = g[M x K] @ W[K x N] + b   (N==2)
__global__ void final_linear(const float* __restrict__ g, const float* __restrict__ W,
                             const float* __restrict__ b, float* out, int M, int K, int N)
{
    int idx = blockIdx.x * blockDim.x + threadIdx.x;
    if (idx >= M * N) return;
    int row = idx / N, col = idx - row * N;
    float acc = b[col];
    for (int k = 0; k < K; ++k) acc += g[row * K + k] * W[k * N + col];
    out[idx] = acc;
}

// ---------------------------------------------------------------------------
static inline int cdiv(int a, int b) { return (a + b - 1) / b; }

static void launch_gemm(const float* A, const float* B, float* C,
                        int M, int N, int K, hipStream_t stream)
{
    const int blocks = (M / 16) * (N / 16);
    switch (K) {
        case 4:   wmma_gemm_k<4>  <<<blocks, 32, 0, stream>>>(A, B, C, M, N); break;
        case 16:  wmma_gemm_k<16> <<<blocks, 32, 0, stream>>>(A, B, C, M, N); break;
        case 32:  wmma_gemm_k<32> <<<blocks, 32, 0, stream>>>(A, B, C, M, N); break;
        case 48:  wmma_gemm_k<48> <<<blocks, 32, 0, stream>>>(A, B, C, M, N); break;
        case 64:  wmma_gemm_k<64> <<<blocks, 32, 0, stream>>>(A, B, C, M, N); break;
        case 96:  wmma_gemm_k<96> <<<blocks, 32, 0, stream>>>(A, B, C, M, N); break;
        case 128: wmma_gemm_k<128><<<blocks, 32, 0, stream>>>(A, B, C, M, N); break;
    }
}

extern "C" void kernel_launch(void* const* d_in, const int* in_sizes, int n_in,
                              void* d_out, int out_size, void* d_ws, size_t ws_size,
                              hipStream_t stream)
{
    const float* x         = (const float*)d_in[0];          // [N,3]
    const int*   edge_idx  = (const int*)d_in[1];            // [2,E]
    const int*   batch_ids = (const int*)d_in[2];            // [N]
    const int E = in_sizes[1] / 2;
    const int N = in_sizes[2];
    const int G = 256;

    const float* convW[5] = {(const float*)d_in[3],  (const float*)d_in[7],
                             (const float*)d_in[11], (const float*)d_in[15],
                             (const float*)d_in[19]};
    const float* bng[5]   = {(const float*)d_in[5],  (const float*)d_in[9],
                             (const float*)d_in[13], (const float*)d_in[17],
                             (const float*)d_in[21]};
    const float* bnb[5]   = {(const float*)d_in[6],  (const float*)d_in[10],
                             (const float*)d_in[14], (const float*)d_in[18],
                             (const float*)d_in[22]};
    const float* linW0 = (const float*)d_in[23];
    const float* linW1 = (const float*)d_in[25];
    const float* linW2 = (const float*)d_in[27];
    const float* linb2 = (const float*)d_in[28];
    const float* mbng0 = (const float*)d_in[29];
    const float* mbnb0 = (const float*)d_in[30];
    const float* mbng1 = (const float*)d_in[31];
    const float* mbnb1 = (const float*)d_in[32];

    const int dims[6] = {4, 16, 32, 48, 64, 96};   // layer 0 padded to K=4
    const int* src = edge_idx;       // edge_index[0]
    const int* dst = edge_idx + E;   // edge_index[1]

    // workspace layout (fp32)
    float* ws = (float*)d_ws;
    size_t o = 0;
    float* bufA  = ws + o; o += (size_t)N * 96;
    float* bufHW = ws + o; o += (size_t)N * 96;
    float* bufC  = ws + o; o += (size_t)N * 96;
    float* xp    = ws + o; o += (size_t)N * 4;     // padded x
    float* w0p   = ws + o; o += 64;                // padded conv_W0
    float* dinv  = ws + o; o += N;
    float* bsum  = ws + o; o += 128;               // bsum & bssq contiguous
    float* bssq  = ws + o; o += 128;
    float* scale = ws + o; o += 128;
    float* shift = ws + o; o += 128;
    float* psum  = ws + o; o += (size_t)G * 96;    // psum & pcnt contiguous
    float* pcnt  = ws + o; o += G;
    float* g0    = ws + o; o += (size_t)G * 128;
    float* g1    = ws + o; o += (size_t)G * 128;
    (void)ws_size; (void)n_in; (void)out_size;

    const int TB = 256;

    // ---- pad layer-0 operands ----
    pad_x<<<cdiv(N, TB), TB, 0, stream>>>(x, xp, N);
    pad_w0<<<1, 64, 0, stream>>>(convW[0], w0p);

    // ---- GCN symmetric normalization: dinv = rsqrt(deg) with self-loop ----
    fill_f32<<<cdiv(N, TB), TB, 0, stream>>>(dinv, 1.0f, N);
    count_deg<<<cdiv(E, TB), TB, 0, stream>>>(dst, dinv, E);
    finalize_dinv<<<cdiv(N, TB), TB, 0, stream>>>(dinv, N);

    // ---- 5 GCNConv + BN + leaky-relu layers ----
    float* pingpong[2] = {bufA, bufC};
    const float* cur = xp;
    for (int l = 0; l < 5; ++l) {
        const int din = dims[l], dout = dims[l + 1];
        const float* W = (l == 0) ? w0p : convW[l];
        float* agg = pingpong[l & 1];
        // hw = cur @ W  (WMMA)
        launch_gemm(cur, W, bufHW, N, dout, din, stream);
        // aggregate: agg = segment_sum(hw[src]*norm, dst)  (+ self loops)
        zero_f32<<<cdiv(N * dout, TB), TB, 0, stream>>>(agg, N * dout);
        edge_scatter4<<<cdiv(E * (dout / 4), TB), TB, 0, stream>>>(
            bufHW, src, dst, dinv, agg, E, dout / 4);
        self_loop_add4<<<cdiv(N * (dout / 4), TB), TB, 0, stream>>>(
            bufHW, dinv, agg, N, dout / 4);
        // BatchNorm (training stats) + leaky relu; conv bias cancels in BN
        zero_f32<<<1, 256, 0, stream>>>(bsum, 256);
        bn_stats<<<512, TB, 0, stream>>>(agg, bsum, bssq, N, dout);
        bn_finalize<<<1, 128, 0, stream>>>(bsum, bssq, bng[l], bnb[l], scale, shift, N, dout);
        bn_apply_leaky<<<cdiv(N * dout, TB), TB, 0, stream>>>(agg, scale, shift, N * dout, dout);
        cur = agg;
    }

    // ---- scatter_mean pool over graphs ----
    zero_f32<<<cdiv(G * 96 + G, TB), TB, 0, stream>>>(psum, G * 96 + G);
    pool_scatter<<<cdiv(N * 96, TB), TB, 0, stream>>>(cur, batch_ids, psum, pcnt, N, 96);
    pool_div<<<cdiv(G * 96, TB), TB, 0, stream>>>(psum, pcnt, G * 96, 96);

    // ---- MLP: 96 -> 128 (BN+leaky) -> 96 (BN+leaky) -> 2 ----
    launch_gemm(psum, linW0, g0, G, 128, 96, stream);
    zero_f32<<<1, 256, 0, stream>>>(bsum, 256);
    bn_stats<<<64, TB, 0, stream>>>(g0, bsum, bssq, G, 128);
    bn_finalize<<<1, 128, 0, stream>>>(bsum, bssq, mbng0, mbnb0, scale, shift, G, 128);
    bn_apply_leaky<<<cdiv(G * 128, TB), TB, 0, stream>>>(g0, scale, shift, G * 128, 128);

    launch_gemm(g0, linW1, g1, G, 96, 128, stream);
    zero_f32<<<1, 256, 0, stream>>>(bsum, 256);
    bn_stats<<<64, TB, 0, stream>>>(g1, bsum, bssq, G, 96);
    bn_finalize<<<1, 128, 0, stream>>>(bsum, bssq, mbng1, mbnb1, scale, shift, G, 96);
    bn_apply_leaky<<<cdiv(G * 96, TB), TB, 0, stream>>>(g1, scale, shift, G * 96, 96);

    final_linear<<<cdiv(G * 2, TB), TB, 0, stream>>>(g1, linW2, linb2, (float*)d_out, G, 96, 2);
}